// phaseDiagram_35459249996555
// MI455X (gfx1250) — compile-verified
//
#include <hip/hip_runtime.h>
#include <stdint.h>

// Problem constants (fixed by setup_inputs in the reference)
#define B_DIM    32
#define L_DIM    65536
#define N_DIM    16
#define EMB_DIM  3
#define DELAY    100
#define OUT_LEN  (L_DIM - (EMB_DIM - 1) * DELAY)   // 65336
#define ROW_F    (EMB_DIM * N_DIM)                 // 48 floats per output row
#define ROW_V4   (ROW_F / 4)                       // 12 float4 per output row

#define BLOCK    192                               // 6 waves; 192 = 16 * 12
#define ROWS_PER_ITER (BLOCK / ROW_V4)             // 16 rows covered per step
#define TILE_T   256                               // output rows per block
#define NITER    (TILE_T / ROWS_PER_ITER)          // 16 store steps
#define HALO     ((EMB_DIM - 1) * DELAY)           // 200
#define SM_IN    (TILE_T + HALO)                   // 456 staged x0 values

// Native clang vector type: works with __builtin_nontemporal_store,
// lowers to a single global_store_b128.
typedef __attribute__((ext_vector_type(4))) float v4f;

// LDS byte offset of a __shared__ object: low 32 bits of the flat shared-
// aperture address are the wave-relative LDS byte address (ISA 10.2).
__device__ __forceinline__ unsigned lds_offset(const void* p) {
    return (unsigned)(uintptr_t)p;
}

// CDNA5 async global -> LDS copy, 4 bytes per active lane (ASYNCcnt-tracked).
__device__ __forceinline__ void async_gload_b32_to_lds(unsigned lds_byte_off,
                                                       const float* gaddr) {
    asm volatile("global_load_async_to_lds_b32 %0, %1, off"
                 :: "v"(lds_byte_off), "v"(gaddr)
                 : "memory");
}

__device__ __forceinline__ void wait_asynccnt0() {
#if __has_builtin(__builtin_amdgcn_s_wait_asynccnt)
    __builtin_amdgcn_s_wait_asynccnt(0);
#else
    asm volatile("s_wait_asynccnt 0" ::: "memory");
#endif
}

__global__ __launch_bounds__(BLOCK)
void delay_embed_tile_kernel(const float* __restrict__ in,
                             float* __restrict__ out) {
    __shared__ float s_x0[SM_IN];

    const int b   = blockIdx.y;
    const int t0  = blockIdx.x * TILE_T;
    const int tid = threadIdx.x;

    // Base of inputs[b, :, 0]; element (b, t, 0) sits at in_b + t*N_DIM.
    const float* in_b = in + (size_t)b * L_DIM * N_DIM;

    // ---- Phase 1: stage x0[b, t0 .. t0+SM_IN) into LDS (async path) ----
    for (int i = tid; i < SM_IN; i += BLOCK) {
        const int t = t0 + i;
        if (t < L_DIM) {
            async_gload_b32_to_lds(lds_offset(&s_x0[i]),
                                   in_b + (size_t)t * N_DIM);
        }
    }
    wait_asynccnt0();
    __syncthreads();

    // ---- Phase 2: expand + stream out ----
    // Thread -> (row0, c) fixed once; iteration k advances row by 16,
    // flat float4 index by 192, LDS byte addr by 64. All strides constant,
    // so the unrolled body is ds_load_b32 + splat + global_store_b128 with
    // immediate offsets.
    const int row0 = tid / ROW_V4;                 // 0..15
    const int c    = tid - row0 * ROW_V4;          // 0..11
    const float* s_base = &s_x0[row0 + (c >> 2) * DELAY];
    v4f* __restrict__ outv =
        (v4f*)(out + ((size_t)b * OUT_LEN + t0) * ROW_F) + tid;

    if (t0 + TILE_T <= OUT_LEN) {
        // Fast path: full tile, no bounds checks.
#pragma unroll
        for (int k = 0; k < NITER; ++k) {
            const float v = s_base[k * ROWS_PER_ITER];
            const v4f val = {v, v, v, v};
            __builtin_nontemporal_store(val, outv + k * BLOCK);
        }
    } else {
        // Tail tile (56 rows per batch).
        const int rows = OUT_LEN - t0;
#pragma unroll
        for (int k = 0; k < NITER; ++k) {
            if (row0 + k * ROWS_PER_ITER < rows) {
                const float v = s_base[k * ROWS_PER_ITER];
                const v4f val = {v, v, v, v};
                __builtin_nontemporal_store(val, outv + k * BLOCK);
            }
        }
    }
}

extern "C" void kernel_launch(void* const* d_in, const int* in_sizes, int n_in,
                              void* d_out, int out_size, void* d_ws, size_t ws_size,
                              hipStream_t stream) {
    (void)in_sizes; (void)n_in; (void)d_ws; (void)ws_size; (void)out_size;
    const float* in  = (const float*)d_in[0];
    float*       out = (float*)d_out;

    dim3 grid((OUT_LEN + TILE_T - 1) / TILE_T, B_DIM);  // 256 x 32
    delay_embed_tile_kernel<<<grid, BLOCK, 0, stream>>>(in, out);
}